// CTCLayer_87462714015866
// MI455X (gfx1250) — compile-verified
//
#include <hip/hip_runtime.h>
#include <hip/hip_bf16.h>
#include <stdint.h>

// ---------------------------------------------------------------------------
// CTC loss forward (log-space alpha recursion), MI455X / gfx1250.
//
// Outputs (concatenated in d_out, float32):
//   [0 .. T*B*C)  : log_tm = log(transpose(y_pred, (1,0,2)))   [T,B,C]
//   [T*B*C]       : mean CTC loss over batch
//
// B=64, T=1024, C=128, L=256, S=2L+1=513.
// ---------------------------------------------------------------------------

#define NEGINF (-1e30f)

static constexpr int Bn = 64;
static constexpr int Tn = 1024;
static constexpr int Cn = 128;
static constexpr int Ln = 256;
static constexpr int Sn = 2 * Ln + 1; // 513

// ---------------------------------------------------------------------------
// CDNA5 async global->LDS copy (ASYNCcnt path). The clang builtin takes
// typed address-space pointers to 16-byte int vectors (per the round-1
// diagnostic). Guarded: falls back to plain load+LDS store if absent.
// ---------------------------------------------------------------------------
#if defined(__HIP_DEVICE_COMPILE__) && __has_builtin(__builtin_amdgcn_global_load_async_to_lds_b128)
#define CTC_ASYNC_LDS 1
#endif

typedef int ctc_v4i __attribute__((vector_size(4 * sizeof(int))));
typedef __attribute__((address_space(1))) ctc_v4i* ctc_gptr;
typedef __attribute__((address_space(3))) ctc_v4i* ctc_lptr;

__device__ __forceinline__ void async_copy16(const float* __restrict__ g,
                                             float* __restrict__ l) {
#ifdef CTC_ASYNC_LDS
  __builtin_amdgcn_global_load_async_to_lds_b128(
      (ctc_gptr)(g), (ctc_lptr)(l), /*offset=*/0, /*cpol=*/0);
#else
  float4 v = *reinterpret_cast<const float4*>(g);
  *reinterpret_cast<float4*>(l) = v;
#endif
}

__device__ __forceinline__ void wait_async0() {
#ifdef CTC_ASYNC_LDS
#if __has_builtin(__builtin_amdgcn_s_wait_asynccnt)
  __builtin_amdgcn_s_wait_asynccnt(0);
#else
  asm volatile("s_wait_asynccnt 0" ::: "memory");
#endif
#endif
}

// ---------------------------------------------------------------------------
// Kernel 1: log_tm[t,b,c] = log(y_pred[b,t,c]); float4-vectorized, both
// sides coalesced along C. 67 MB of traffic -> HBM-bound (~2.9 us @ 23.3TB/s).
// ---------------------------------------------------------------------------
__global__ void k_log_transpose(const float* __restrict__ y_pred,
                                float* __restrict__ log_tm) {
  const int i = blockIdx.x * blockDim.x + threadIdx.x; // float4 index of output
  const int c  = (i & 31) << 2;  // C/4 = 32 float4 per row
  const int bt = i >> 5;         // = t*B + b
  const int b  = bt & (Bn - 1);
  const int t  = bt >> 6;        // /B
  const float4 v =
      *reinterpret_cast<const float4*>(y_pred + ((size_t)b * Tn + t) * Cn + c);
  float4 o;
  o.x = __logf(v.x); o.y = __logf(v.y); o.z = __logf(v.z); o.w = __logf(v.w);
  *reinterpret_cast<float4*>(log_tm + ((size_t)t * Bn + b) * Cn + c) = o;
}

// ---------------------------------------------------------------------------
// Kernel 2: per-batch alpha recursion. One block per batch element.
//   - 544 threads = 17 waves. Threads 0..512 each own one extended state s.
//   - Wave 16 (tids 512..543, full 32-lane EXEC) async-prefetches the next
//     time step's 128-float probability row into LDS (b128 per lane = 512B).
//   - alpha kept in LDS, double buffered, with a 2-entry NEGINF pad in front
//     so a1/a2 reads need no edge branches.
//   - Exactly one s_wait_asynccnt + one barrier per time step; the prefetch
//     of row t+1 overlaps the logsumexp compute of step t.
//   - Freeze past input_length == just stop the loop at clipped il.
// ---------------------------------------------------------------------------
__global__ void __launch_bounds__(544)
k_ctc_alpha(const float* __restrict__ y_pred, const int* __restrict__ y_true,
            const int* __restrict__ input_length,
            const int* __restrict__ label_length, float* __restrict__ loss_b) {
  __shared__ float alpha[2][Sn + 2]; // [*][0..1] = NEGINF pad, state s at 2+s
  __shared__ float row[2][Cn];       // double-buffered probability row

  const int b = blockIdx.x;
  const int tid = threadIdx.x;
  const int s = tid;

  int il = input_length[b]; il = il < 1 ? 1 : (il > Tn ? Tn : il);
  int ll = label_length[b]; ll = ll < 1 ? 1 : (ll > Ln ? Ln : ll);

  // Per-thread extended-label class and skip-transition flag (registers).
  int cls = 0;        // blank
  bool skip = false;  // s-2 -> s transition allowed
  if (s < Sn && (s & 1)) {
    cls = y_true[b * Ln + (s >> 1)];
    // s==1 reads the NEGINF pad for a2 anyway, so skip=false there is fine.
    skip = (s >= 3) && (cls != y_true[b * Ln + ((s - 3) >> 1)]);
  }
  if (tid < 2) { alpha[0][tid] = NEGINF; alpha[1][tid] = NEGINF; }

  const float* rowbase = y_pred + (size_t)b * Tn * Cn;
  const bool loader = (tid >= 512);   // wave 16, all 32 lanes
  const int lane = tid - 512;

  // Row 0: issue + wait + publish.
  if (loader) {
    async_copy16(rowbase + lane * 4, &row[0][lane * 4]);
    wait_async0();
  }
  __syncthreads();

  // alpha0: only s=0 (blank) and s=1 (first label) are live.
  if (s < Sn) {
    const float lp = __logf(row[0][cls]);
    alpha[0][2 + s] = (s < 2) ? lp : NEGINF;
  }
  // Prefetch row 1 (overlaps nothing yet, but keeps steady state uniform).
  if (loader && il > 1)
    async_copy16(rowbase + Cn + lane * 4, &row[1][lane * 4]);

  for (int t = 1; t < il; ++t) {
    const int rb = t & 1;
    if (loader) wait_async0();  // row t landed in LDS (only one outstanding)
    __syncthreads();            // publish row t; step t-1 compute all done
    if (loader && (t + 1 < il)) // overwrite row t-1's buffer: safe after bar
      async_copy16(rowbase + (size_t)(t + 1) * Cn + lane * 4,
                   &row[rb ^ 1][lane * 4]);
    if (s < Sn) {
      const float* cur = alpha[(t + 1) & 1];
      const float a0 = cur[2 + s];
      const float a1 = cur[1 + s];
      const float a2 = skip ? cur[s] : NEGINF;
      const float m = fmaxf(a0, fmaxf(a1, a2));
      const float r =
          m + __logf(__expf(a0 - m) + __expf(a1 - m) + __expf(a2 - m));
      alpha[t & 1][2 + s] = r + __logf(row[rb][cls]);
    }
    // next iteration's top barrier orders this write vs. its reads
  }
  __syncthreads();

  if (tid == 0) {
    const float* cur = alpha[(il - 1) & 1];
    const float e1 = cur[2 + 2 * ll - 1];
    const float e2 = cur[2 + 2 * ll];
    const float m = fmaxf(e1, e2);
    loss_b[b] = -(m + __logf(__expf(e1 - m) + __expf(e2 - m)));
  }
}

// ---------------------------------------------------------------------------
// Kernel 3: deterministic mean over 64 per-batch losses.
// ---------------------------------------------------------------------------
__global__ void k_mean_loss(const float* __restrict__ loss_b,
                            float* __restrict__ out) {
  __shared__ float sm[Bn];
  const int tid = threadIdx.x;
  sm[tid] = loss_b[tid];
  __syncthreads();
  if (tid == 0) {
    float acc = 0.0f;
    for (int i = 0; i < Bn; ++i) acc += sm[i];
    *out = acc * (1.0f / (float)Bn);
  }
}

// ---------------------------------------------------------------------------
extern "C" void kernel_launch(void* const* d_in, const int* in_sizes, int n_in,
                              void* d_out, int out_size, void* d_ws,
                              size_t ws_size, hipStream_t stream) {
  const int*   y_true       = (const int*)d_in[0];
  const float* y_pred       = (const float*)d_in[1];
  const int*   input_length = (const int*)d_in[2];
  const int*   label_length = (const int*)d_in[3];

  float* log_tm   = (float*)d_out;                        // T*B*C floats
  float* loss_out = log_tm + (size_t)Tn * Bn * Cn;        // final element
  float* loss_b   = (float*)d_ws;                         // 64-float scratch

  // 1) log + transpose: (T*B*C)/4 float4 elements.
  const int n4 = (Tn * Bn * Cn) / 4;       // 2,097,152
  k_log_transpose<<<n4 / 256, 256, 0, stream>>>(y_pred, log_tm);

  // 2) per-batch alpha recursion (64 WGPs busy, LDS-resident state).
  k_ctc_alpha<<<Bn, 544, 0, stream>>>(y_pred, y_true, input_length,
                                      label_length, loss_b);

  // 3) mean over batch.
  k_mean_loss<<<1, Bn, 0, stream>>>(loss_b, loss_out);
}